// WGCNSingleChannel_85048942395829
// MI455X (gfx1250) — compile-verified
//
#include <hip/hip_runtime.h>
#include <math.h>

#define N_NODES 100000
#define IN_F    256
#define OUT_F   64
#define D_CH    4
#define N_EDGES 800000

typedef __attribute__((ext_vector_type(2))) float v2f;
typedef __attribute__((ext_vector_type(8))) float v8f;

// ---------------------------------------------------------------------------
// monotone uint key encoding for float atomic-max
// ---------------------------------------------------------------------------
__device__ __forceinline__ unsigned f2key(float f) {
    unsigned u = __float_as_uint(f);
    return (u & 0x80000000u) ? ~u : (u | 0x80000000u);
}
__device__ __forceinline__ float key2f(unsigned k) {
    unsigned u = (k & 0x80000000u) ? (k & 0x7FFFFFFFu) : ~k;
    return __uint_as_float(u);
}
#define NEG_INF_KEY 0x007FFFFFu  // f2key(-inf)

__device__ __forceinline__ float leaky02(float x) {
    return x > 0.0f ? x : 0.2f * x;
}

// ---------------------------------------------------------------------------
// Wh[n][64] = (feature[n][:256] @ W[256][64]) * norm[n]
// One block = 16 rows, 4 waves; each wave owns a 16-column stripe and does a
// 16x16 tile via 64x V_WMMA_F32_16X16X4_F32 over K=256.
// A layout (16x4 f32): lane L<16 -> {A[L][k], A[L][k+1]}, lane L+16 -> {A[L][k+2], A[L][k+3]}
// B layout (4x16 f32): lane L<16 -> {B[k][L], B[k+1][L]},  lane L+16 -> {B[k+2][L], B[k+3][L]}
// C/D layout: vgpr r, lane L -> D[r + 8*(L>=16)][L&15]
// ---------------------------------------------------------------------------
__global__ __launch_bounds__(128) void gemm_wmma_kernel(
    const float* __restrict__ feat,   // [N][256]
    const float* __restrict__ norm,   // [N]
    const float* __restrict__ W,      // [256][64] (one head)
    float* __restrict__ Wh)           // [N][64]
{
    __shared__ float Atile[16][IN_F];   // 16 KB
    __shared__ float norms[16];

    const int row0 = blockIdx.x * 16;
    const int tid  = threadIdx.x;

    // cooperative stage of the 16x256 feature tile into LDS (float4 chunks)
    {
        const float4* srcp = (const float4*)(feat + (size_t)row0 * IN_F);
        float4* dstp = (float4*)(&Atile[0][0]);
        #pragma unroll
        for (int i = 0; i < (16 * IN_F / 4) / 128; ++i)
            dstp[tid + i * 128] = srcp[tid + i * 128];
        if (tid < 16) norms[tid] = norm[row0 + tid];
        // hint-prefetch next block's feature tile (gfx1250 global_prefetch_b8)
        if (row0 + 16 < N_NODES)
            __builtin_prefetch(feat + (size_t)(row0 + 16) * IN_F + tid * 8, 0, 0);
    }
    __syncthreads();

    const int wave  = tid >> 5;
    const int lane  = tid & 31;
    const int half  = lane >> 4;       // 0: K lo pair, 1: K hi pair
    const int mn    = lane & 15;       // A's M index == B's N index for this lane
    const int ncol0 = wave * 16;

    v8f c = {};
    #pragma unroll 4
    for (int k = 0; k < IN_F; k += 4) {
        const int kk = k + 2 * half;
        v2f a;
        a.x = Atile[mn][kk];
        a.y = Atile[mn][kk + 1];
        v2f b;
        b.x = W[(size_t)kk * OUT_F + ncol0 + mn];
        b.y = W[(size_t)(kk + 1) * OUT_F + ncol0 + mn];
        c = __builtin_amdgcn_wmma_f32_16x16x4_f32(
                /*neg_a=*/false, a, /*neg_b=*/false, b,
                /*c_mod=*/(short)0, c, /*reuse_a=*/false, /*reuse_b=*/false);
    }

    #pragma unroll
    for (int r = 0; r < 8; ++r) {
        const int m = r + 8 * half;
        Wh[(size_t)(row0 + m) * OUT_F + ncol0 + mn] = c[r] * norms[m];
    }
}

// ---------------------------------------------------------------------------
// p_src[n] = Wh[n] . att_w[0:64] ; p_dst[n] = Wh[n] . att_w[64:128]
// ---------------------------------------------------------------------------
__global__ void attn_proj_kernel(const float* __restrict__ Wh,
                                 const float* __restrict__ att_w,
                                 float* __restrict__ p_src,
                                 float* __restrict__ p_dst)
{
    const int n = blockIdx.x * blockDim.x + threadIdx.x;
    if (n >= N_NODES) return;
    const float4* row = (const float4*)(Wh + (size_t)n * OUT_F);
    const float4* as  = (const float4*)att_w;
    const float4* ad  = (const float4*)(att_w + OUT_F);
    float s = 0.0f, d = 0.0f;
    #pragma unroll
    for (int i = 0; i < OUT_F / 4; ++i) {
        float4 v = row[i];
        float4 a0 = as[i];
        float4 a1 = ad[i];
        s = fmaf(v.x, a0.x, fmaf(v.y, a0.y, fmaf(v.z, a0.z, fmaf(v.w, a0.w, s))));
        d = fmaf(v.x, a1.x, fmaf(v.y, a1.y, fmaf(v.z, a1.z, fmaf(v.w, a1.w, d))));
    }
    p_src[n] = s;
    p_dst[n] = d;
}

// ---------------------------------------------------------------------------
// per-head node state init: z = 0, max-key = key(-inf)
// ---------------------------------------------------------------------------
__global__ void init_nodes_kernel(unsigned* __restrict__ mk, float* __restrict__ z)
{
    const int n = blockIdx.x * blockDim.x + threadIdx.x;
    if (n >= N_NODES) return;
    mk[n] = NEG_INF_KEY;
    z[n]  = 0.0f;
}

// ---------------------------------------------------------------------------
// segment max of e = leaky_relu(p_src[s] + p_dst[d]) over dst
// ---------------------------------------------------------------------------
__global__ void edge_max_kernel(const float* __restrict__ p_src,
                                const float* __restrict__ p_dst,
                                const int* __restrict__ src,
                                const int* __restrict__ dst,
                                unsigned* __restrict__ mk)
{
    const int e = blockIdx.x * blockDim.x + threadIdx.x;
    if (e >= N_EDGES) return;
    const int s = src[e];
    const int d = dst[e];
    const float ev = leaky02(p_src[s] + p_dst[d]);
    atomicMax(&mk[d], f2key(ev));
}

// ---------------------------------------------------------------------------
// segment sum of w = exp(e - m[dst]) over dst   (e recomputed, not stored)
// ---------------------------------------------------------------------------
__global__ void edge_sum_kernel(const float* __restrict__ p_src,
                                const float* __restrict__ p_dst,
                                const int* __restrict__ src,
                                const int* __restrict__ dst,
                                const unsigned* __restrict__ mk,
                                float* __restrict__ z)
{
    const int e = blockIdx.x * blockDim.x + threadIdx.x;
    if (e >= N_EDGES) return;
    const int s = src[e];
    const int d = dst[e];
    const float ev = leaky02(p_src[s] + p_dst[d]);
    const float m  = key2f(mk[d]);
    atomicAdd(&z[d], __expf(ev - m));
}

// ---------------------------------------------------------------------------
// out[dst][head*64 + f] += (w/z) * Wh[src][f] : one wave32 per edge, float2/lane
// ---------------------------------------------------------------------------
__global__ __launch_bounds__(256) void edge_agg_kernel(
    const float* __restrict__ Wh,
    const float* __restrict__ p_src,
    const float* __restrict__ p_dst,
    const int* __restrict__ src,
    const int* __restrict__ dst,
    const unsigned* __restrict__ mk,
    const float* __restrict__ z,
    float* __restrict__ out,          // [N][256]
    int head)
{
    const long long gid = (long long)blockIdx.x * blockDim.x + threadIdx.x;
    const int e    = (int)(gid >> 5);
    const int lane = (int)(gid & 31);
    if (e >= N_EDGES) return;

    const int s = src[e];
    const int d = dst[e];
    const float ev = leaky02(p_src[s] + p_dst[d]);
    const float m  = key2f(mk[d]);
    const float a  = __expf(ev - m) / z[d];

    const float2 wh = ((const float2*)(Wh + (size_t)s * OUT_F))[lane];
    float* o = out + (size_t)d * (D_CH * OUT_F) + head * OUT_F + 2 * lane;
    atomicAdd(o + 0, a * wh.x);
    atomicAdd(o + 1, a * wh.y);
}

// ---------------------------------------------------------------------------
// output zero / finalize (out = relu(out * norm[n]))
// ---------------------------------------------------------------------------
__global__ void zero_out_kernel(float* __restrict__ out)
{
    const long long i = (long long)blockIdx.x * blockDim.x + threadIdx.x;
    if (i < (long long)N_NODES * (D_CH * OUT_F)) out[i] = 0.0f;
}

__global__ void finalize_kernel(float* __restrict__ out, const float* __restrict__ norm)
{
    const long long i = (long long)blockIdx.x * blockDim.x + threadIdx.x;
    if (i >= (long long)N_NODES * (D_CH * OUT_F)) return;
    const int n = (int)(i / (D_CH * OUT_F));
    const float v = out[i] * norm[n];
    out[i] = v > 0.0f ? v : 0.0f;
}

// ---------------------------------------------------------------------------
// host launcher
// ---------------------------------------------------------------------------
extern "C" void kernel_launch(void* const* d_in, const int* in_sizes, int n_in,
                              void* d_out, int out_size, void* d_ws, size_t ws_size,
                              hipStream_t stream)
{
    (void)in_sizes; (void)n_in; (void)out_size; (void)ws_size;

    const float* feat  = (const float*)d_in[0];   // [N][256]
    const float* norm  = (const float*)d_in[1];   // [N]
    const float* W     = (const float*)d_in[2];   // [4][256][64]
    const float* att_w = (const float*)d_in[3];   // [128]
    const int*   src   = (const int*)d_in[4];     // [4][E]
    const int*   dst   = (const int*)d_in[5];     // [4][E]
    float* out = (float*)d_out;                   // [N][256]

    // workspace slab (~27.2 MB), reused per head
    float*    Wh    = (float*)d_ws;                               // N*64
    float*    p_src = Wh + (size_t)N_NODES * OUT_F;               // N
    float*    p_dst = p_src + N_NODES;                            // N
    unsigned* mk    = (unsigned*)(p_dst + N_NODES);               // N
    float*    z     = (float*)(mk + N_NODES);                     // N

    const int TPB = 256;
    const long long out_elems = (long long)N_NODES * (D_CH * OUT_F);
    const int out_blocks  = (int)((out_elems + TPB - 1) / TPB);
    const int node_blocks = (N_NODES + TPB - 1) / TPB;
    const int edge_blocks = (N_EDGES + TPB - 1) / TPB;
    const int agg_blocks  = (int)(((long long)N_EDGES * 32 + TPB - 1) / TPB);
    const int gemm_blocks = N_NODES / 16;   // 100000 divisible by 16

    zero_out_kernel<<<out_blocks, TPB, 0, stream>>>(out);

    for (int h = 0; h < D_CH; ++h) {
        gemm_wmma_kernel<<<gemm_blocks, 128, 0, stream>>>(
            feat, norm, W + (size_t)h * IN_F * OUT_F, Wh);
        attn_proj_kernel<<<node_blocks, TPB, 0, stream>>>(Wh, att_w, p_src, p_dst);
        init_nodes_kernel<<<node_blocks, TPB, 0, stream>>>(mk, z);
        edge_max_kernel<<<edge_blocks, TPB, 0, stream>>>(
            p_src, p_dst, src + (size_t)h * N_EDGES, dst + (size_t)h * N_EDGES, mk);
        edge_sum_kernel<<<edge_blocks, TPB, 0, stream>>>(
            p_src, p_dst, src + (size_t)h * N_EDGES, dst + (size_t)h * N_EDGES, mk, z);
        edge_agg_kernel<<<agg_blocks, TPB, 0, stream>>>(
            Wh, p_src, p_dst, src + (size_t)h * N_EDGES, dst + (size_t)h * N_EDGES,
            mk, z, out, h);
    }

    finalize_kernel<<<out_blocks, TPB, 0, stream>>>(out, norm);
}